// DiffBPF_34643206209868
// MI455X (gfx1250) — compile-verified
//
#include <hip/hip_runtime.h>
#include <hip/hip_bf16.h>

#define NPART   16384
#define TSTEPS  4096
#define NTHR    1024
#define PPT     16            // particles per thread

typedef float v2f  __attribute__((ext_vector_type(2)));
typedef float v8f  __attribute__((ext_vector_type(8)));
typedef unsigned int ux4 __attribute__((ext_vector_type(4)));
typedef int ix4 __attribute__((ext_vector_type(4)));
typedef int ix8 __attribute__((ext_vector_type(8)));

// ---------------------------------------------------------------------------
// TDM: DMA one eps row (16384 f32, 64KB) from global memory into LDS.
// D# built per CDNA5 ISA ch.8 (group0: count/lds_addr/global_addr/type=2,
// group1: data_size=4B, tensor_dim0=16384, tile_dim0=16384, 1 row).
// Issued by wave 0 only; completion tracked with TENSORcnt.
// ---------------------------------------------------------------------------
__device__ __forceinline__ void tdm_load_row(const float* gsrc, unsigned lds_byte)
{
    unsigned long long ga = (unsigned long long)(const void*)gsrc;
    ux4 g0;
    g0[0] = 1u;                                   // count=1, user descriptor
    g0[1] = lds_byte;                             // LDS byte address
    g0[2] = (unsigned)ga;                         // global_addr[31:0]
    g0[3] = ((unsigned)(ga >> 32) & 0x01FFFFFFu)  // global_addr[56:32]
          | (2u << 30);                           // type=2 ("image")
    ix8 g1;
    g1[0] = (int)(2u << 16);                      // data_size=2 -> 4 bytes
    g1[1] = (int)((16384u & 0xFFFFu) << 16);      // tensor_dim0[15:0] @ [63:48]
    g1[2] = (int)(1u << 16);                      // tensor_dim0 hi=0, tensor_dim1=1
    g1[3] = (int)(16384u << 16);                  // tile_dim0=16384 @ [127:112]
    g1[4] = 1;                                    // tile_dim1=1, tile_dim2=0
    g1[5] = 16384;                                // tensor_dim0_stride lo32
    g1[6] = (int)((16384u & 0xFFFFu) << 16);      // tensor_dim1_stride[15:0]
    g1[7] = 0;
    ix4 z4 = {0, 0, 0, 0};
#if defined(__clang_major__) && __clang_major__ >= 23
    ix8 z8 = {0, 0, 0, 0, 0, 0, 0, 0};
    __builtin_amdgcn_tensor_load_to_lds(g0, g1, z4, z4, z8, 0);
#else
    __builtin_amdgcn_tensor_load_to_lds(g0, g1, z4, z4, 0);
#endif
}

// ---------------------------------------------------------------------------
// Persistent single-workgroup BPF. 1024 threads = 32 wave32 waves on one WGP.
// LDS layout (dynamic, ~200KB of the WGP's 320KB):
//   h[16384] | eps ping[16384] | eps pong[16384] | ttot[1024] | tscan[1024]
//   | wmax[32] | shp[32] | svp[32] | coltot[64] | tiletot[4]
// ---------------------------------------------------------------------------
__global__ void __launch_bounds__(NTHR, 1)
bpf_persistent_kernel(const float* __restrict__ mu_raw_p,
                      const float* __restrict__ rho_raw_p,
                      const float* __restrict__ lsz_p,
                      const float* __restrict__ nu_raw_p,
                      const float* __restrict__ y_seq,
                      const float* __restrict__ h_init,
                      const float* __restrict__ eps_seq,
                      const float* __restrict__ u_seq,
                      float* __restrict__ out)
{
    extern __shared__ float smem[];
    float* hS    = smem;                 // 16384
    float* eps0  = smem + 16384;         // 16384
    float* eps1  = smem + 32768;         // 16384
    float* ttot  = smem + 49152;         // 1024 per-thread weight totals
    float* tscan = smem + 50176;         // 1024 scanned totals
    float* wmaxS = smem + 51200;         // 32 wave maxes
    float* shpS  = smem + 51232;         // 32 wave sums of e*h
    float* svpS  = smem + 51264;         // 32 wave sums of e*exp(h/2)
    float* colS  = smem + 51296;         // 4 tiles * 16 column totals
    float* tiletotS = smem + 51360;      // 4 tile totals

    const int tid  = threadIdx.x;
    const int lane = tid & 31;
    const int wid  = tid >> 5;

    // ---- constrained parameters (uniform; every thread computes the same) ----
    const float mu  = mu_raw_p[0];
    const float rho = 1.0f / (1.0f + __expf(-rho_raw_p[0]));
    const float sz  = log1pf(__expf(lsz_p[0]));                 // softplus
    const float nu  = 2.0f + log1pf(__expf(nu_raw_p[0]));
    const float halfnup1 = 0.5f * (nu + 1.0f);
    const float lconst = lgammaf(halfnup1) - lgammaf(0.5f * nu)
                       - 0.5f * __logf(nu * 3.14159265358979f);
    const float logN = __logf((float)NPART);

    // ---- stationary init: copy h_init into LDS (each thread its own slots) ----
    #pragma unroll
    for (int j = 0; j < PPT; ++j)
        hS[tid * PPT + j] = h_init[tid * PPT + j];

    // prime the eps pipeline: TDM row 0 -> eps0 (wave 0 only; EXEC-independent)
    if (tid < 32)
        tdm_load_row(eps_seq, (unsigned)(unsigned long long)(void*)eps0);

    for (int t = 0; t < TSTEPS; ++t) {
        float* epsBuf  = (t & 1) ? eps1 : eps0;
        float* epsNext = (t & 1) ? eps0 : eps1;

        // wait for this step's eps row, then launch the DMA for the next one.
        // Safe: all waves finished reading epsNext (== buffer of step t-1)
        // before wave 0 could pass the barriers of step t-1.
        if (tid < 32) {
            __builtin_amdgcn_s_wait_tensorcnt(0);
            if (t + 1 < TSTEPS)
                tdm_load_row(eps_seq + (size_t)(t + 1) * NPART,
                             (unsigned)(unsigned long long)(void*)epsNext);
        }
        __syncthreads();   // eps[t] visible; resampled h from step t-1 visible

        const float yt = y_seq[t];
        const float ut = u_seq[t];
        const float y2 = yt * yt;

        // ---- phase A: predict + log-weights, per-thread local max ----
        float hv[PPT], lw[PPT];
        float lmax = -3.0e38f;
        #pragma unroll
        for (int j = 0; j < PPT; ++j) {
            const int i = tid * PPT + j;
            float ho = hS[i];
            float hn = fmaf(rho, ho - mu, mu) + sz * epsBuf[i];
            hv[j] = hn;
            float lwj = lconst - 0.5f * hn
                      - halfnup1 * log1pf(y2 / (nu * __expf(hn) + 1e-8f));
            lw[j] = lwj;
            lmax = fmaxf(lmax, lwj);
        }
        #pragma unroll
        for (int o = 16; o >= 1; o >>= 1)
            lmax = fmaxf(lmax, __shfl_xor(lmax, o, 32));
        if (lane == 0) wmaxS[wid] = lmax;
        __syncthreads();

        float gmax = -3.0e38f;
        #pragma unroll
        for (int k = 0; k < 32; ++k) gmax = fmaxf(gmax, wmaxS[k]);

        // ---- phase B: e = exp(lw - gmax), local inclusive prefix + moments ----
        float lp[PPT];
        float run = 0.f, sh = 0.f, sv = 0.f;
        #pragma unroll
        for (int j = 0; j < PPT; ++j) {
            float e = __expf(lw[j] - gmax);
            run += e; lp[j] = run;
            sh += e * hv[j];
            sv += e * __expf(0.5f * hv[j]);
        }
        const float s = run;
        float rsh = sh, rsv = sv;
        #pragma unroll
        for (int o = 16; o >= 1; o >>= 1) {
            rsh += __shfl_xor(rsh, o, 32);
            rsv += __shfl_xor(rsv, o, 32);
        }
        if (lane == 0) { shpS[wid] = rsh; svpS[wid] = rsv; }
        ttot[tid] = s;
        __syncthreads();

        // ---- phase C: WMMA prefix scan of the 1024 thread totals ----
        // 4 tiles of 16x16 (column-major element order = thread order),
        // waves 0..3 in parallel. P = L@X via 4 chained v_wmma_f32_16x16x4_f32,
        // then per-column offsets from column totals (row 15 of P).
        if (wid < 4) {
            const int base = wid << 8;
            const int m  = lane & 15;              // A row / B column / D column
            const int hi = (lane >> 4) << 1;       // K pairing: 0 | 2 (ISA 7.12.2)
            v8f acc = {};
            #pragma unroll
            for (int kk = 0; kk < 4; ++kk) {
                const int ka = 4 * kk + hi;
                v2f a, b;
                a.x = (ka     <= m) ? 1.f : 0.f;   // L[m, ka]
                a.y = (ka + 1 <= m) ? 1.f : 0.f;   // L[m, ka+1]
                b.x = ttot[base + m * 16 + ka];    // X[ka,   col=m]
                b.y = ttot[base + m * 16 + ka + 1];// X[ka+1, col=m]
                acc = __builtin_amdgcn_wmma_f32_16x16x4_f32(
                          false, a, false, b, (short)0, acc, false, false);
            }
            // column totals live at M=15 -> VGPR 7, lanes 16..31
            if (lane >= 16) colS[(wid << 4) + (lane - 16)] = acc[7];
            float off = 0.f, tot = 0.f;
            const int c = lane & 15;
            #pragma unroll
            for (int cc = 0; cc < 16; ++cc) {      // in-order DS pipe: RAW safe
                float vv = colS[(wid << 4) + cc];
                tot += vv;
                off += (cc < c) ? vv : 0.f;
            }
            const int mrow = (lane < 16) ? 0 : 8;
            #pragma unroll
            for (int r = 0; r < 8; ++r)
                tscan[base + c * 16 + (mrow + r)] = acc[r] + off;
            if (lane == 0) tiletotS[wid] = tot;
        }
        __syncthreads();

        // ---- phase D: normalization + O(N) systematic-resample scatter ----
        const float sumE = tiletotS[0] + tiletotS[1] + tiletotS[2] + tiletotS[3];
        const float inv  = 1.0f / sumE;
        const int  mytile = tid >> 8;
        float tb = 0.f;
        #pragma unroll
        for (int k = 0; k < 4; ++k) tb += (k < mytile) ? tiletotS[k] : 0.f;
        const float exc   = tb + tscan[tid] - s;     // exclusive prefix (raw e)
        const float scale = (float)NPART * inv;      // raw-cum -> pos*N space

        // particle i owns cum-interval (a,b]; it fills output slots k with
        // a*N < u + k <= b*N  (searchsorted-left semantics, cs[-1]=1 forced).
        float aprev = exc * scale;
        #pragma unroll
        for (int j = 0; j < PPT; ++j) {
            float bj = (exc + lp[j]) * scale;
            if (tid == NTHR - 1 && j == PPT - 1) bj = (float)NPART; // cs[-1]=1
            int klo = (int)floorf(aprev - ut) + 1;
            int khi = (int)floorf(bj - ut);
            klo = klo < 0 ? 0 : klo;
            khi = khi > NPART - 1 ? NPART - 1 : khi;
            for (int k = klo; k <= khi; ++k) hS[k] = hv[j];
            aprev = bj;
        }

        // ---- outputs ----
        if (tid == 0) {
            float shTot = 0.f, svTot = 0.f;
            #pragma unroll
            for (int k = 0; k < 32; ++k) { shTot += shpS[k]; svTot += svpS[k]; }
            out[t]              = gmax + __logf(sumE) - logN;  // log_lik
            out[TSTEPS + t]     = shTot * inv;                 // h_mean
            out[2 * TSTEPS + t] = svTot * inv;                 // vol
        }
        // next iteration's top barrier orders scatter writes vs. phase-A reads
    }
}

extern "C" void kernel_launch(void* const* d_in, const int* in_sizes, int n_in,
                              void* d_out, int out_size, void* d_ws, size_t ws_size,
                              hipStream_t stream) {
    (void)in_sizes; (void)n_in; (void)out_size; (void)d_ws; (void)ws_size;
    const float* mu_raw  = (const float*)d_in[0];
    const float* rho_raw = (const float*)d_in[1];
    const float* lsz     = (const float*)d_in[2];
    const float* nu_raw  = (const float*)d_in[3];
    const float* y_seq   = (const float*)d_in[4];
    const float* h_init  = (const float*)d_in[5];
    const float* eps_seq = (const float*)d_in[6];
    const float* u_seq   = (const float*)d_in[7];
    float* out = (float*)d_out;

    const size_t smem_bytes = (size_t)51364 * sizeof(float);   // ~200.6 KB < 320 KB
    (void)hipFuncSetAttribute((const void*)bpf_persistent_kernel,
                              hipFuncAttributeMaxDynamicSharedMemorySize,
                              (int)smem_bytes);
    bpf_persistent_kernel<<<dim3(1), dim3(NTHR), smem_bytes, stream>>>(
        mu_raw, rho_raw, lsz, nu_raw, y_seq, h_init, eps_seq, u_seq, out);
}